// ProtDiGCNEncoderDecoder_minibatch_15444702396417
// MI455X (gfx1250) — compile-verified
//
#include <hip/hip_runtime.h>
#include <hip/hip_bf16.h>
#include <math.h>

typedef float v2f __attribute__((ext_vector_type(2)));
typedef float v8f __attribute__((ext_vector_type(8)));

// Non-returning relaxed agent-scope f32 atomic add: lowers to
// global_atomic_add_f32 (no TH_ATOMIC_RETURN, STOREcnt-tracked).
__device__ __forceinline__ void atomic_add_f32(float* p, float v)
{
    (void)__hip_atomic_fetch_add(p, v, __ATOMIC_RELAXED, __HIP_MEMORY_SCOPE_AGENT);
}

// ---------------------------------------------------------------------------
// WMMA f32 GEMM: C[M x Nc] = A[M x K] * B[K x Nc] (+ bias per column).
// One 16x16 output tile per wave, V_WMMA_F32_16X16X4_F32, K stepped by 4.
// M % 16 == 0, Nc % 16 == 0, K % 4 == 0 (all true for this workload).
// ---------------------------------------------------------------------------
__global__ __launch_bounds__(256) void wmma_gemm_f32(
    const float* __restrict__ A, const float* __restrict__ B,
    float* __restrict__ C, const float* __restrict__ bias,
    int M, int Nc, int K)
{
    const int ntn   = Nc >> 4;
    const int total = (M >> 4) * ntn;
    const int w     = blockIdx.x * 8 + (threadIdx.x >> 5);
    if (w >= total) return;                  // whole-wave uniform guard

    const int mt = w / ntn;
    const int nt = w - mt * ntn;
    const int m0 = mt << 4;
    const int n0 = nt << 4;

    const int lane = threadIdx.x & 31;
    const int wq   = lane & 15;              // row (A) / col (B,C) within tile
    const int hi   = lane >> 4;              // K-half selector

    v8f acc = {};

    const float* ap = A + (size_t)(m0 + wq) * K + (hi << 1);
    const float* bp = B + (size_t)(hi << 1) * Nc + n0 + wq;

    for (int k = 0; k < K; k += 4) {
        v2f a; a.x = ap[0]; a.y = ap[1];       // A[m][k+2*hi], A[m][k+2*hi+1]
        v2f b; b.x = bp[0]; b.y = bp[Nc];      // B[k+2*hi][n], B[k+2*hi+1][n]
        ap += 4;
        bp += (size_t)4 * Nc;
        acc = __builtin_amdgcn_wmma_f32_16x16x4_f32(
            false, a, false, b, (short)0, acc, false, false);
    }

    const float bv = bias ? bias[n0 + wq] : 0.0f;
    float* cp = C + (size_t)(m0 + (hi << 3)) * Nc + n0 + wq;
#pragma unroll
    for (int r = 0; r < 8; ++r) {            // VGPR r -> rows m0+r / m0+r+8
        cp[0] = acc[r] + bv;
        cp += Nc;
    }
}

// ---------------------------------------------------------------------------
// Degree count: deg[dst] += 1 per edge (accumulated into dinv buffer).
// ---------------------------------------------------------------------------
__global__ void degree_kernel(const int* __restrict__ dst, float* __restrict__ deg, int nE)
{
    int e = blockIdx.x * blockDim.x + threadIdx.x;
    if (e < nE) atomic_add_f32(&deg[dst[e]], 1.0f);
}

__global__ void finish_dinv(float* __restrict__ dinv, int n)
{
    int i = blockIdx.x * blockDim.x + threadIdx.x;
    if (i < n) dinv[i] = rsqrtf(dinv[i] + 1.0f);   // deg >= 1
}

// ---------------------------------------------------------------------------
// Edge aggregation: agg[dst] += h[src] * dinv[src]*dinv[dst].
// Each thread handles 4 channels (float4 gather + 4 f32 atomics).
// ---------------------------------------------------------------------------
template <int C>
__global__ __launch_bounds__(256) void edge_agg(
    const int* __restrict__ src, const int* __restrict__ dst,
    const float* __restrict__ dinv, const float* __restrict__ h,
    float* __restrict__ agg, int nE)
{
    constexpr int TPE = C / 4;               // threads per edge
    constexpr int EPB = 256 / TPE;           // edges per block
    const int sub = threadIdx.x % TPE;
    const int e   = blockIdx.x * EPB + threadIdx.x / TPE;
    if (e >= nE) return;

    const int s = src[e];
    const int d = dst[e];
    const float coef = dinv[s] * dinv[d];

    const float4 v = *((const float4*)(h + (size_t)s * C) + sub);
    float* ap = agg + (size_t)d * C + sub * 4;
    atomic_add_f32(ap + 0, v.x * coef);
    atomic_add_f32(ap + 1, v.y * coef);
    atomic_add_f32(ap + 2, v.z * coef);
    atomic_add_f32(ap + 3, v.w * coef);
}

// ---------------------------------------------------------------------------
// Layer-1 epilogue: h = relu(agg + h0 * dinv^2 + b1), in place on agg (C=128).
// One float4 group per thread.
// ---------------------------------------------------------------------------
__global__ void combine1_relu(float* __restrict__ agg, const float* __restrict__ h0,
                              const float* __restrict__ dinv, const float* __restrict__ b,
                              int nGroups)      // nGroups = N * (128/4)
{
    int t = blockIdx.x * blockDim.x + threadIdx.x;
    if (t >= nGroups) return;
    const int i  = t >> 5;                    // node
    const int c4 = t & 31;                    // float4 channel group
    const float di = dinv[i];
    const float sl = di * di;
    float4 a  = ((const float4*)agg)[t];
    float4 h  = ((const float4*)h0)[t];
    float4 bb = ((const float4*)b)[c4];
    float4 r;
    r.x = fmaxf(fmaf(h.x, sl, a.x) + bb.x, 0.0f);
    r.y = fmaxf(fmaf(h.y, sl, a.y) + bb.y, 0.0f);
    r.z = fmaxf(fmaf(h.z, sl, a.z) + bb.z, 0.0f);
    r.w = fmaxf(fmaf(h.w, sl, a.w) + bb.w, 0.0f);
    ((float4*)agg)[t] = r;
}

// ---------------------------------------------------------------------------
// Layer-2 epilogue + row L2-normalize (C=64). One wave per row, 2 ch/lane.
// ---------------------------------------------------------------------------
__global__ __launch_bounds__(256) void combine2_norm(
    const float* __restrict__ agg2, const float* __restrict__ h2pre,
    const float* __restrict__ dinv, const float* __restrict__ b2,
    float* __restrict__ emb, int N)
{
    const int lane = threadIdx.x & 31;
    const int row  = blockIdx.x * 8 + (threadIdx.x >> 5);
    if (row >= N) return;
    const float di = dinv[row];
    const float sl = di * di;
    const size_t base = (size_t)row * 64 + lane * 2;
    const float2 a  = *(const float2*)(agg2 + base);
    const float2 h  = *(const float2*)(h2pre + base);
    const float2 bb = *(const float2*)(b2 + lane * 2);
    const float vx = fmaf(h.x, sl, a.x) + bb.x;
    const float vy = fmaf(h.y, sl, a.y) + bb.y;
    float ss = vx * vx + vy * vy;
#pragma unroll
    for (int m = 16; m >= 1; m >>= 1) ss += __shfl_xor(ss, m, 32);
    const float inv = 1.0f / (sqrtf(ss) + 1e-12f);
    float2 o; o.x = vx * inv; o.y = vy * inv;
    *(float2*)(emb + base) = o;
}

// ---------------------------------------------------------------------------
// Row log-softmax over 256 logits. One wave per row, 8 ch/lane.
// ---------------------------------------------------------------------------
__global__ __launch_bounds__(256) void log_softmax_rows(
    const float* __restrict__ logits, float* __restrict__ out, int N)
{
    const int lane = threadIdx.x & 31;
    const int row  = blockIdx.x * 8 + (threadIdx.x >> 5);
    if (row >= N) return;
    const float* lp = logits + (size_t)row * 256;
    float v[8];
    float mx = -INFINITY;
#pragma unroll
    for (int j = 0; j < 8; ++j) { v[j] = lp[lane + j * 32]; mx = fmaxf(mx, v[j]); }
#pragma unroll
    for (int m = 16; m >= 1; m >>= 1) mx = fmaxf(mx, __shfl_xor(mx, m, 32));
    float s = 0.0f;
#pragma unroll
    for (int j = 0; j < 8; ++j) s += expf(v[j] - mx);
#pragma unroll
    for (int m = 16; m >= 1; m >>= 1) s += __shfl_xor(s, m, 32);
    const float lse = mx + logf(s);
    float* op = out + (size_t)row * 256;
#pragma unroll
    for (int j = 0; j < 8; ++j) op[lane + j * 32] = v[j] - lse;
}

// ---------------------------------------------------------------------------
// Host-side orchestration.
// Inputs: 0:x[N,64] 1:edge_index[2,E] 2:W1[64,128] 3:b1[128]
//         4:W2[128,64] 5:b2[64] 6:Wd[64,256] 7:bd[256]
// Output: log_softmax[N,256] ++ final_embeddings[N,64]
// ---------------------------------------------------------------------------
extern "C" void kernel_launch(void* const* d_in, const int* in_sizes, int n_in,
                              void* d_out, int out_size, void* d_ws, size_t ws_size,
                              hipStream_t stream)
{
    const float* x   = (const float*)d_in[0];
    const int*   ei  = (const int*)  d_in[1];
    const float* W1  = (const float*)d_in[2];
    const float* b1  = (const float*)d_in[3];
    const float* W2  = (const float*)d_in[4];
    const float* b2  = (const float*)d_in[5];
    const float* Wd  = (const float*)d_in[6];
    const float* bd  = (const float*)d_in[7];

    const int N  = in_sizes[0] / 64;       // 100000 nodes
    const int nE = in_sizes[1] / 2;        // 3.2M edges
    const int* src = ei;
    const int* dst = ei + nE;

    float* ws     = (float*)d_ws;
    float* h0     = ws;                        // N*128
    float* agg1   = ws + (size_t)N * 128;      // N*128 (becomes h after relu)
    float* h2pre  = ws + (size_t)N * 256;      // N*64
    float* agg2   = ws + (size_t)N * 320;      // N*64
    float* dinv   = ws + (size_t)N * 384;      // N     (deg -> dinv)
    float* logits = ws;                        // N*256, reuses h0+agg1 (dead by then)

    float* out_lsm = (float*)d_out;                       // N*256
    float* out_emb = (float*)d_out + (size_t)N * 256;     // N*64

    // Zero accumulators (graph-capturable async memsets).
    hipMemsetAsync(dinv, 0, (size_t)N * sizeof(float), stream);
    hipMemsetAsync(agg1, 0, (size_t)N * 128 * sizeof(float), stream);
    hipMemsetAsync(agg2, 0, (size_t)N * 64 * sizeof(float), stream);

    // Degree -> dinv = rsqrt(deg+1).
    degree_kernel<<<(nE + 255) / 256, 256, 0, stream>>>(dst, dinv, nE);
    finish_dinv<<<(N + 255) / 256, 256, 0, stream>>>(dinv, N);

    // Layer 1: h0 = x @ W1 (WMMA), scatter-add over edges, relu-combine.
    {
        const int tiles = (N / 16) * (128 / 16);
        wmma_gemm_f32<<<(tiles + 7) / 8, 256, 0, stream>>>(x, W1, h0, nullptr, N, 128, 64);
        const int epb = 256 / (128 / 4);   // 8 edges per block
        edge_agg<128><<<(nE + epb - 1) / epb, 256, 0, stream>>>(src, dst, dinv, h0, agg1, nE);
        const int groups = N * (128 / 4);
        combine1_relu<<<(groups + 255) / 256, 256, 0, stream>>>(agg1, h0, dinv, b1, groups);
    }

    // Layer 2: h2pre = h @ W2 (WMMA), scatter-add, combine + row-normalize.
    {
        const int tiles = (N / 16) * (64 / 16);
        wmma_gemm_f32<<<(tiles + 7) / 8, 256, 0, stream>>>(agg1, W2, h2pre, nullptr, N, 64, 128);
        const int epb = 256 / (64 / 4);    // 16 edges per block
        edge_agg<64><<<(nE + epb - 1) / epb, 256, 0, stream>>>(src, dst, dinv, h2pre, agg2, nE);
        combine2_norm<<<(N + 7) / 8, 256, 0, stream>>>(agg2, h2pre, dinv, b2, out_emb, N);
    }

    // Decoder: logits = emb @ Wd + bd (WMMA), then row log-softmax.
    {
        const int tiles = (N / 16) * (256 / 16);
        wmma_gemm_f32<<<(tiles + 7) / 8, 256, 0, stream>>>(out_emb, Wd, logits, bd, N, 256, 64);
        log_softmax_rows<<<(N + 7) / 8, 256, 0, stream>>>(logits, out_lsm, N);
    }
}